// GCNClient_48936857370858
// MI455X (gfx1250) — compile-verified
//
#include <hip/hip_runtime.h>

#define N_NODES 100000
#define DIM 128
#define MTILES (N_NODES / 16)   // 6250, exact

typedef __attribute__((ext_vector_type(2))) float v2f;
typedef __attribute__((ext_vector_type(8))) float v8f;

// ---------------- degree / normalization ----------------
__global__ void k_init_deg(float* deg) {
    int i = blockIdx.x * blockDim.x + threadIdx.x;
    if (i < N_NODES) deg[i] = 1.0f;                 // self-loop contribution
}

__global__ void k_count_deg(const int* __restrict__ dst, long long E, float* deg) {
    long long e = (long long)blockIdx.x * blockDim.x + threadIdx.x;
    if (e < E) atomicAdd(&deg[dst[e]], 1.0f);
}

__global__ void k_dinv(float* deg) {
    int i = blockIdx.x * blockDim.x + threadIdx.x;
    if (i < N_NODES) deg[i] = rsqrtf(deg[i]);       // deg >= 1 always
}

// ---------------- zero fill (float4) ----------------
__global__ void k_zero(float* p, long long n4) {
    long long i = (long long)blockIdx.x * blockDim.x + threadIdx.x;
    if (i < n4) ((float4*)p)[i] = make_float4(0.f, 0.f, 0.f, 0.f);
}

// ---------------- GEMM: H[N,128] = X[N,128] @ W[128,128], fp32 WMMA ----------------
// One wave computes a 16x128 output strip: 8 accumulator tiles of 16x16,
// K-loop in steps of 4 using V_WMMA_F32_16X16X4_F32.
__global__ __launch_bounds__(256) void k_gemm(const float* __restrict__ X,
                                              const float* __restrict__ W,
                                              float* __restrict__ H) {
    int wave = (blockIdx.x * blockDim.x + threadIdx.x) >> 5;   // wave-uniform
    int lane = threadIdx.x & 31;
    if (wave >= MTILES) return;                                 // whole-wave exit

    const int m0   = wave * 16;
    const int half = lane >> 4;          // 0: lanes 0-15, 1: lanes 16-31
    const int l16  = lane & 15;
    const int koff = half * 2;           // A/B K offset for upper half-wave

    v8f acc[8];
#pragma unroll
    for (int n = 0; n < 8; ++n) acc[n] = (v8f){0,0,0,0,0,0,0,0};

    const float* xrow = X + (size_t)(m0 + l16) * DIM;

    for (int k0 = 0; k0 < DIM; k0 += 4) {
        // A tile 16x4: lane<16 holds K=k0+{0,1}, lane>=16 holds K=k0+{2,3}, M=l16
        v2f a;
        a.x = xrow[k0 + koff + 0];
        a.y = xrow[k0 + koff + 1];
#pragma unroll
        for (int n = 0; n < 8; ++n) {
            // B tile 4x16: VGPR v holds K=k0+koff+v, N = n*16+l16
            v2f b;
            b.x = W[(size_t)(k0 + koff + 0) * DIM + n * 16 + l16];
            b.y = W[(size_t)(k0 + koff + 1) * DIM + n * 16 + l16];
            acc[n] = __builtin_amdgcn_wmma_f32_16x16x4_f32(
                false, a, false, b, (short)0, acc[n], false, false);
        }
    }

    // C/D layout: VGPR v -> row m0 + v + (lane<16 ? 0 : 8), col = n*16 + l16
    const int rbase = m0 + half * 8;
#pragma unroll
    for (int n = 0; n < 8; ++n) {
#pragma unroll
        for (int v = 0; v < 8; ++v) {
            H[(size_t)(rbase + v) * DIM + n * 16 + l16] = acc[n][v];
        }
    }
}

// ---------------- edge aggregation: out[dst] += dinv[src]*dinv[dst] * h[src] ----------------
// One wave per edge; each lane moves 4 consecutive floats (32 lanes x 4 = 128).
__global__ void k_aggregate(const float* __restrict__ H,
                            const int* __restrict__ src,
                            const int* __restrict__ dst,
                            const float* __restrict__ dinv,
                            float* __restrict__ out, long long E) {
    long long t = (long long)blockIdx.x * blockDim.x + threadIdx.x;
    long long e = t >> 5;
    int lane = threadIdx.x & 31;
    if (e >= E) return;

    int s = src[e];
    int d = dst[e];
    float nrm = dinv[s] * dinv[d];

    float4 v = ((const float4*)(H + (size_t)s * DIM))[lane];
    float* op = out + (size_t)d * DIM + lane * 4;
    atomicAdd(op + 0, v.x * nrm);
    atomicAdd(op + 1, v.y * nrm);
    atomicAdd(op + 2, v.z * nrm);
    atomicAdd(op + 3, v.w * nrm);
}

// ---------------- fused: out = relu(agg + dinv[i]^2 * h + bias) ----------------
__global__ void k_bias_relu_self(const float* __restrict__ agg,
                                 const float* __restrict__ H,
                                 const float* __restrict__ dinv,
                                 const float* __restrict__ bias,
                                 float* __restrict__ out) {
    long long idx = (long long)blockIdx.x * blockDim.x + threadIdx.x;  // float4 granularity
    const long long n4 = (long long)N_NODES * DIM / 4;
    if (idx >= n4) return;
    int i  = (int)(idx >> 5);      // node (32 float4 per row)
    int f4 = (int)(idx & 31);

    float di = dinv[i];
    float sn = di * di;            // self-loop norm
    float4 a  = ((const float4*)agg)[idx];
    float4 h  = ((const float4*)H)[idx];
    float4 bb = ((const float4*)bias)[f4];
    float4 r;
    r.x = fmaxf(fmaf(sn, h.x, a.x) + bb.x, 0.f);
    r.y = fmaxf(fmaf(sn, h.y, a.y) + bb.y, 0.f);
    r.z = fmaxf(fmaf(sn, h.z, a.z) + bb.z, 0.f);
    r.w = fmaxf(fmaf(sn, h.w, a.w) + bb.w, 0.f);
    ((float4*)out)[idx] = r;
}

extern "C" void kernel_launch(void* const* d_in, const int* in_sizes, int n_in,
                              void* d_out, int out_size, void* d_ws, size_t ws_size,
                              hipStream_t stream) {
    const float* x   = (const float*)d_in[0];
    const int*   ei  = (const int*)  d_in[1];   // edge_index [2, E]
    const float* W1  = (const float*)d_in[2];
    const float* b1  = (const float*)d_in[3];
    const float* W2  = (const float*)d_in[4];
    const float* b2  = (const float*)d_in[5];
    float* out = (float*)d_out;

    const long long E = (long long)in_sizes[1] / 2;
    const int* src = ei;
    const int* dst = ei + E;

    // workspace layout: dinv[N] | bufA[N*D] | bufB[N*D]
    float* dinv = (float*)d_ws;
    float* bufA = dinv + N_NODES;                  // 400000 B offset, 16B aligned
    float* bufB = bufA + (size_t)N_NODES * DIM;

    const long long nd  = (long long)N_NODES * DIM;
    const long long nd4 = nd / 4;

    dim3 blk(256);
    const int gN    = (N_NODES + 255) / 256;
    const int gE    = (int)((E + 255) / 256);
    const int gZ    = (int)((nd4 + 255) / 256);
    const int gAgg  = (int)((E * 32 + 255) / 256);
    const int gGemm = (MTILES + 7) / 8;            // 8 waves per 256-thread block

    // normalization: deg (with self-loop) -> dinv
    k_init_deg<<<gN, blk, 0, stream>>>(dinv);
    k_count_deg<<<gE, blk, 0, stream>>>(dst, E, dinv);
    k_dinv<<<gN, blk, 0, stream>>>(dinv);

    // ---- layer 1 ----
    k_gemm<<<gGemm, blk, 0, stream>>>(x, W1, bufA);            // h1 = x @ W1
    k_zero<<<gZ, blk, 0, stream>>>(bufB, nd4);
    k_aggregate<<<gAgg, blk, 0, stream>>>(bufA, src, dst, dinv, bufB, E);
    k_bias_relu_self<<<gZ, blk, 0, stream>>>(bufB, bufA, dinv, b1, bufA);  // x2 in bufA

    // ---- layer 2 ----
    k_gemm<<<gGemm, blk, 0, stream>>>(bufA, W2, bufB);         // h2 = x2 @ W2
    k_zero<<<gZ, blk, 0, stream>>>(out, nd4);
    k_aggregate<<<gAgg, blk, 0, stream>>>(bufB, src, dst, dinv, out, E);
    k_bias_relu_self<<<gZ, blk, 0, stream>>>(out, bufB, dinv, b2, out);
}